// GridPooling_84052509982724
// MI455X (gfx1250) — compile-verified
//
#include <hip/hip_runtime.h>
#include <hip/hip_bf16.h>
#include <stdint.h>

#define GRID_SIZE 0.04f
#define AXIS_CAP  32
#define USE_ASYNC_LDS 1

// Order-preserving float <-> uint encoding (for atomic min on floats)
__device__ __forceinline__ unsigned encf(float f) {
  unsigned u = __float_as_uint(f);
  return (u & 0x80000000u) ? ~u : (u | 0x80000000u);
}
__device__ __forceinline__ float decf(unsigned u) {
  unsigned b = (u & 0x80000000u) ? (u & 0x7FFFFFFFu) : ~u;
  return __uint_as_float(b);
}

// ---------------- K0: workspace init ----------------
__global__ void k0_init(unsigned* __restrict__ minEnc, unsigned* __restrict__ counts,
                        unsigned* __restrict__ cursor, float* __restrict__ coordSum,
                        unsigned* __restrict__ nvoxP, int NKEYS, int B) {
  int i = blockIdx.x * blockDim.x + threadIdx.x;
  if (i < NKEYS) {
    counts[i] = 0u;
    cursor[i] = 0u;
    coordSum[(size_t)i * 3 + 0] = 0.0f;
    coordSum[(size_t)i * 3 + 1] = 0.0f;
    coordSum[(size_t)i * 3 + 2] = 0.0f;
  }
  if (i < B * 3) minEnc[i] = 0xFFFFFFFFu;
  if (i == 0) nvoxP[0] = 0u;
}

// ---------------- K1: per-batch coordinate min ----------------
__global__ void k1_batchmin(const float* __restrict__ coord, const int* __restrict__ offs,
                            unsigned* __restrict__ minEnc, int N, int B) {
  __shared__ unsigned smin[48];   // B*3, B <= 16
  const int t = threadIdx.x;
  if (t < B * 3) smin[t] = 0xFFFFFFFFu;
  __syncthreads();
  const int i = blockIdx.x * blockDim.x + t;
  if (i < N) {
    int b = 0;
    for (int k = 0; k < B - 1; ++k) b += (i >= offs[k]) ? 1 : 0;
    #pragma unroll
    for (int k = 0; k < 3; ++k) {
      float c = coord[(size_t)i * 3 + k];
      atomicMin(&smin[b * 3 + k], encf(c));     // ds_min_u32
    }
  }
  __syncthreads();
  if (t < B * 3) atomicMin(&minEnc[t], smin[t]);
}

// ---------------- K2: voxelize + histogram + coord sums ----------------
__global__ void k2_voxel(const float* __restrict__ coord, const int* __restrict__ offs,
                         const unsigned* __restrict__ minEnc,
                         unsigned* __restrict__ counts, float* __restrict__ coordSum,
                         unsigned* __restrict__ pointKey, int N, int B) {
  __shared__ float sstart[48];
  const int t = threadIdx.x;
  if (t < B * 3) sstart[t] = decf(minEnc[t]);
  __syncthreads();
  const int i = blockIdx.x * blockDim.x + t;
  if (i >= N) return;
  int b = 0;
  for (int k = 0; k < B - 1; ++k) b += (i >= offs[k]) ? 1 : 0;
  float c[3]; int v[3];
  #pragma unroll
  for (int k = 0; k < 3; ++k) {
    c[k] = coord[(size_t)i * 3 + k];
    int q = (int)floorf((c[k] - sstart[b * 3 + k]) / GRID_SIZE);
    v[k] = min(max(q, 0), AXIS_CAP - 1);
  }
  unsigned key = (((unsigned)b * AXIS_CAP + (unsigned)v[0]) * AXIS_CAP + (unsigned)v[1]) * AXIS_CAP
                 + (unsigned)v[2];
  pointKey[i] = key;
  atomicAdd(&counts[key], 1u);
  atomicAdd(&coordSum[(size_t)key * 3 + 0], c[0]);
  atomicAdd(&coordSum[(size_t)key * 3 + 1], c[1]);
  atomicAdd(&coordSum[(size_t)key * 3 + 2], c[2]);
}

// ---------------- K3: single-workgroup scan over bins ----------------
// Produces binStart (excl. scan of counts), keyRank (rank -> key),
// per-batch voxel counts -> offset_out tail, n_vox.
__global__ __launch_bounds__(1024)
void k3_scan(const unsigned* __restrict__ counts,
             unsigned* __restrict__ binStart, unsigned* __restrict__ keyRank,
             unsigned* __restrict__ nvoxP, float* __restrict__ tailOut,
             int NKEYS, int B) {
  __shared__ unsigned long long sc[1024];
  __shared__ unsigned pb[16];
  const int t = threadIdx.x;
  if (t < 16) pb[t] = 0u;
  const int CH = (NKEYS + 1023) >> 10;
  const int k0 = t * CH;
  const int k1 = min(k0 + CH, NKEYS);
  unsigned cs = 0u, fs = 0u;
  for (int k = k0; k < k1; ++k) { unsigned cc = counts[k]; cs += cc; fs += (cc > 0u); }
  unsigned long long v = ((unsigned long long)fs << 32) | (unsigned long long)cs;
  sc[t] = v;
  __syncthreads();
  for (int off = 1; off < 1024; off <<= 1) {
    unsigned long long u = (t >= off) ? sc[t - off] : 0ull;
    __syncthreads();
    sc[t] += u;
    __syncthreads();
  }
  const unsigned long long excl = sc[t] - v;
  const unsigned long long tot  = sc[1023];
  unsigned runC = (unsigned)excl;
  unsigned runF = (unsigned)(excl >> 32);
  for (int k = k0; k < k1; ++k) {
    unsigned cc = counts[k];
    binStart[k] = runC;
    if (cc > 0u) {
      keyRank[runF] = (unsigned)k;
      ++runF;
      atomicAdd(&pb[(unsigned)k >> 15], 1u);   // key/32768 = batch
    }
    runC += cc;
  }
  __syncthreads();
  if (t == 0) {
    unsigned nv = (unsigned)(tot >> 32);
    nvoxP[0] = nv;
    unsigned cum = 0u;
    for (int bb = 0; bb < B; ++bb) { cum += pb[bb]; tailOut[bb] = (float)cum; }
    tailOut[B] = (float)nv;   // n_vox
  }
}

// ---------------- K4: counting-sort scatter of point indices ----------------
__global__ void k4_scatter(const unsigned* __restrict__ pointKey,
                           const unsigned* __restrict__ binStart,
                           unsigned* __restrict__ cursor,
                           unsigned* __restrict__ sortedIdx, int N) {
  const int i = blockIdx.x * blockDim.x + threadIdx.x;
  if (i >= N) return;
  unsigned key = pointKey[i];
  unsigned pos = atomicAdd(&cursor[key], 1u);
  sortedIdx[binStart[key] + pos] = (unsigned)i;
}

// ---------------- K5: per-segment reduce (1 wave32 / output row) ----------------
// feat max via async global->LDS double buffer; coord mean; zero padding rows.
__global__ __launch_bounds__(256)
void k5_out(const float* __restrict__ feat,
            const unsigned* __restrict__ counts,
            const unsigned* __restrict__ binStart,
            const unsigned* __restrict__ keyRank,
            const float* __restrict__ coordSum,
            const unsigned* __restrict__ sortedIdx,
            const unsigned* __restrict__ nvoxP,
            float* __restrict__ coordOut, float* __restrict__ featOut,
            int N, int C) {
  __shared__ float buf[8][2][64];            // 8 waves x double buffer x 64ch
  const int w    = threadIdx.x >> 5;
  const int lane = threadIdx.x & 31;
  const int s    = blockIdx.x * 8 + w;
  if (s >= N) return;
  const unsigned nvox = nvoxP[0];
  float2* frow = (float2*)(featOut + (size_t)s * C + (lane << 1));
  if ((unsigned)s >= nvox) {                 // padded rows -> zeros
    if (lane < 3) coordOut[(size_t)s * 3 + lane] = 0.0f;
    *frow = make_float2(0.0f, 0.0f);
    return;
  }
  const unsigned key  = keyRank[s];
  const unsigned cnt  = counts[key];
  const unsigned base = binStart[key];
  if (lane < 3)
    coordOut[(size_t)s * 3 + lane] = coordSum[(size_t)key * 3 + lane] / (float)cnt;

  float mx = -INFINITY, my = -INFINITY;
#if USE_ASYNC_LDS
  auto issue = [&](unsigned j, int bsel) {
    unsigned idx = sortedIdx[base + j];
    const float* gp = feat + (size_t)idx * (size_t)C + (lane << 1);
    unsigned long long ga = (unsigned long long)(uintptr_t)gp;
    unsigned lds = (unsigned)(uintptr_t)&buf[w][bsel][lane << 1];
    // wait for our previous LDS read of this buffer before async overwrite
    asm volatile("s_wait_dscnt 0x0\n\t"
                 "global_load_async_to_lds_b64 %0, %1, off"
                 :: "v"(lds), "v"(ga) : "memory");
  };
  issue(0u, 0);
  for (unsigned j = 0; j < cnt; ++j) {
    if (j + 1 < cnt) {
      issue(j + 1, (int)((j + 1) & 1u));
      asm volatile("s_wait_asynccnt 0x1" ::: "memory");   // oldest load done
    } else {
      asm volatile("s_wait_asynccnt 0x0" ::: "memory");
    }
    float a = buf[w][j & 1u][(lane << 1) + 0];
    float b = buf[w][j & 1u][(lane << 1) + 1];
    mx = fmaxf(mx, a);
    my = fmaxf(my, b);
  }
#else
  for (unsigned j = 0; j < cnt; ++j) {
    unsigned idx = sortedIdx[base + j];
    const float2 v = *(const float2*)(feat + (size_t)idx * (size_t)C + (lane << 1));
    mx = fmaxf(mx, v.x);
    my = fmaxf(my, v.y);
  }
#endif
  *frow = make_float2(mx, my);
}

extern "C" void kernel_launch(void* const* d_in, const int* in_sizes, int n_in,
                              void* d_out, int out_size, void* d_ws, size_t ws_size,
                              hipStream_t stream) {
  const float* coord = (const float*)d_in[0];
  const float* feat  = (const float*)d_in[1];
  const int*   offs  = (const int*)d_in[2];

  const int N = in_sizes[0] / 3;
  const int B = in_sizes[2];
  const int C = (N > 0) ? (in_sizes[1] / N) : 64;    // = 64
  const int NKEYS = B * AXIS_CAP * AXIS_CAP * AXIS_CAP;
  if (N <= 0) return;

  // workspace layout (32-bit words)
  unsigned* w32      = (unsigned*)d_ws;
  unsigned* minEnc   = w32;                                  // 64 (uses B*3)
  unsigned* counts   = w32 + 64;                             // NKEYS
  unsigned* cursor   = counts + NKEYS;                       // NKEYS
  unsigned* binStart = cursor + NKEYS;                       // NKEYS
  unsigned* keyRank  = binStart + NKEYS;                     // NKEYS
  float*    coordSum = (float*)(keyRank + NKEYS);            // 3*NKEYS
  unsigned* nvoxP    = (unsigned*)(coordSum + (size_t)3 * NKEYS); // 64
  unsigned* pointKey = nvoxP + 64;                           // N
  unsigned* sortedIdx= pointKey + N;                         // N

  float* out      = (float*)d_out;
  float* coordOut = out;
  float* featOut  = out + (size_t)N * 3;
  float* tailOut  = out + (size_t)N * 3 + (size_t)N * C;     // offset_out[B], n_vox

  const int T = 256;
  dim3 blk(T);

  k0_init<<<dim3((NKEYS + T - 1) / T), blk, 0, stream>>>(minEnc, counts, cursor,
                                                         coordSum, nvoxP, NKEYS, B);
  k1_batchmin<<<dim3((N + T - 1) / T), blk, 0, stream>>>(coord, offs, minEnc, N, B);
  k2_voxel<<<dim3((N + T - 1) / T), blk, 0, stream>>>(coord, offs, minEnc, counts,
                                                      coordSum, pointKey, N, B);
  k3_scan<<<dim3(1), dim3(1024), 0, stream>>>(counts, binStart, keyRank, nvoxP,
                                              tailOut, NKEYS, B);
  k4_scatter<<<dim3((N + T - 1) / T), blk, 0, stream>>>(pointKey, binStart, cursor,
                                                        sortedIdx, N);
  k5_out<<<dim3((N + 7) / 8), blk, 0, stream>>>(feat, counts, binStart, keyRank,
                                                coordSum, sortedIdx, nvoxP,
                                                coordOut, featOut, N, C);
}